// FeatureIPMProjector_63668595196325
// MI455X (gfx1250) — compile-verified
//
#include <hip/hip_runtime.h>
#include <hip/hip_bf16.h>
#include <stdint.h>

// ---------------- problem constants (from reference) ----------------
#define BATCH  2
#define NCAM   6
#define CHN    256
#define HF     48
#define WF     96
#define HFWF   (HF * WF)          // 4608 floats per channel slice
#define BEVH   188
#define BEVW   126
#define NPTS   (BEVH * BEVW)      // 23688 BEV points
// dx = 75.2/188 = 0.4 -> xc = 0.2 + 0.4*i ; dy = 50.4/126 = 0.4 -> yc = -25.0 + 0.4*j

// ---------------- tiling ----------------
#define CH          8                       // channels staged in LDS per WG (8*4608*4 = 144 KB < 320 KB)
#define THREADS     256
#define PPT         4                       // points per thread
#define PTS_PER_BLK (THREADS * PPT)         // 1024
#define SLAB_FLOATS (CH * HFWF)             // 36864
#define SLAB_BYTES  (SLAB_FLOATS * 4)       // 147456

// ---------------- feature detection ----------------
#if defined(__has_builtin)
#  if __has_builtin(__builtin_amdgcn_global_load_async_to_lds_b128)
#    define USE_ASYNC_LDS 1
#  endif
#  if __has_builtin(__builtin_amdgcn_wmma_f32_16x16x4_f32)
#    define USE_WMMA_PROJ 1
#  endif
#  if __has_builtin(__builtin_amdgcn_s_wait_asynccnt)
#    define HAVE_WAIT_ASYNC 1
#  endif
#endif

typedef __attribute__((ext_vector_type(2))) float v2f;
typedef __attribute__((ext_vector_type(8))) float v8f;
typedef int v4i __attribute__((ext_vector_type(4)));

// async-copy builtin wants exact pointee type: v4i in AS(1)/AS(3)
#define ASYNC_GPTR(x) ((__attribute__((address_space(1))) v4i*)(x))
#define ASYNC_SPTR(x) ((__attribute__((address_space(3))) v4i*)(x))

__device__ __forceinline__ void wait_async_zero() {
#if defined(USE_ASYNC_LDS)
#  if defined(HAVE_WAIT_ASYNC)
  __builtin_amdgcn_s_wait_asynccnt(0);
#  else
  asm volatile("s_wait_asynccnt 0" ::: "memory");
#  endif
#endif
}

// Scalar projection of BEV point p for camera (K 3x3, M 4x4). Produces bilinear
// sample coords (ix,iy) in feature-map units; invalid points map to 142.5
// (all four taps out of range -> zero contribution, matching the reference).
__device__ __forceinline__ void project_point(const float* __restrict__ K,
                                              const float* __restrict__ M,
                                              int p, float& ix, float& iy) {
  int bi = p / BEVW;
  int bj = p - bi * BEVW;
  float x = 0.2f + 0.4f * (float)bi;
  float y = -25.0f + 0.4f * (float)bj;
  // P_cam = M @ [x,y,0,1]
  float pc0 = M[0] * x + M[1] * y + M[3];
  float pc1 = M[4] * x + M[5] * y + M[7];
  float pc2 = M[8] * x + M[9] * y + M[11];
  // p_img = K @ P_cam[:3]
  float u = K[0] * pc0 + K[1] * pc1 + K[2] * pc2;
  float v = K[3] * pc0 + K[4] * pc1 + K[5] * pc2;
  float w = K[6] * pc0 + K[7] * pc1 + K[8] * pc2;
  float z  = fmaxf(w, 1e-4f);
  float uf = u / (8.0f * z);   // IMG_W/Wf == IMG_H/Hf == 8
  float vf = v / (8.0f * z);
  bool valid = (pc2 > 0.1f) && (uf >= 0.0f) && (uf <= (float)(WF - 1)) &&
               (vf >= 0.0f) && (vf <= (float)(HF - 1));
  ix = valid ? uf : 142.5f;
  iy = valid ? vf : 142.5f;
}

// ---------------------------------------------------------------------------
// Kernel 1: projection via V_WMMA_F32_16X16X4_F32.
//   D = A(16x4) x B(4x16):  A = G (4x4 fused projection, zero-padded rows),
//   B columns = 16 BEV points [x,y,0,1]. Lane n (n<16) then holds
//   d[0..3] = {u', v', w', Zc} of point n — no cross-lane traffic needed.
// One wave projects 16 points; block of 256 = 8 waves = 128 points.
// ---------------------------------------------------------------------------
__global__ __launch_bounds__(THREADS)
void FeatureIPM_proj_kernel(const float* __restrict__ intr,
                            const float* __restrict__ c2c,
                            float* __restrict__ proj) {
  const int n    = blockIdx.y;
  const int b    = blockIdx.z;
  const int tid  = threadIdx.x;
  const int wave = tid >> 5;
  const int lane = tid & 31;
  const int m    = lane & 15;
  const bool hi  = lane >= 16;
  const int p    = blockIdx.x * 128 + wave * 16 + m;
  const int pc   = (p < NPTS) ? p : (NPTS - 1);

  const float* K = intr + (size_t)(b * NCAM + n) * 9;
  const float* M = c2c  + (size_t)(b * NCAM + n) * 16;

  // G rows 0..2 = K(3x3) @ M(3x4);  row 3 = M row 2 (camera-frame Z)
  float G[4][4];
#pragma unroll
  for (int r = 0; r < 3; ++r)
#pragma unroll
    for (int c = 0; c < 4; ++c)
      G[r][c] = K[r * 3 + 0] * M[c] + K[r * 3 + 1] * M[4 + c] + K[r * 3 + 2] * M[8 + c];
#pragma unroll
  for (int c = 0; c < 4; ++c) G[3][c] = M[8 + c];

  float u, v, w, Zc;
#if defined(USE_WMMA_PROJ)
  // A-matrix: lane m<16 holds row m {K=0,1}; lane m+16 holds row m {K=2,3}.
  float a0 = 0.0f, a1 = 0.0f;
#pragma unroll
  for (int r = 0; r < 4; ++r) {
    if (m == r) {
      a0 = hi ? G[r][2] : G[r][0];
      a1 = hi ? G[r][3] : G[r][1];
    }
  }
  v2f a; a.x = a0; a.y = a1;

  // B-matrix: column n = point n; lanes 0..15 carry {K=0(x), K=1(y)},
  // lanes 16..31 carry {K=2(z=0), K=3(w=1)}.
  int bi = pc / BEVW, bj = pc - bi * BEVW;
  float xc = 0.2f + 0.4f * (float)bi;
  float yc = -25.0f + 0.4f * (float)bj;
  v2f bb; bb.x = hi ? 0.0f : xc; bb.y = hi ? 1.0f : yc;

  v8f d = {};
  d = __builtin_amdgcn_wmma_f32_16x16x4_f32(false, a, false, bb, (short)0, d,
                                            false, false);
  u = d[0]; v = d[1]; w = d[2]; Zc = d[3];
#else
  int bi = pc / BEVW, bj = pc - bi * BEVW;
  float xc = 0.2f + 0.4f * (float)bi;
  float yc = -25.0f + 0.4f * (float)bj;
  u  = G[0][0] * xc + G[0][1] * yc + G[0][3];
  v  = G[1][0] * xc + G[1][1] * yc + G[1][3];
  w  = G[2][0] * xc + G[2][1] * yc + G[2][3];
  Zc = G[3][0] * xc + G[3][1] * yc + G[3][3];
#endif

  if (!hi && p < NPTS) {
    float z  = fmaxf(w, 1e-4f);
    float uf = u / (8.0f * z);
    float vf = v / (8.0f * z);
    bool valid = (Zc > 0.1f) && (uf >= 0.0f) && (uf <= (float)(WF - 1)) &&
                 (vf >= 0.0f) && (vf <= (float)(HF - 1));
    float* o = proj + ((size_t)(b * NCAM + n) * NPTS + p) * 2;
    o[0] = valid ? uf : 142.5f;
    o[1] = valid ? vf : 142.5f;
  }
}

// ---------------------------------------------------------------------------
// Kernel 2: per-WG, stage an 8-channel contiguous feature slab (144 KB) into
// LDS with async b128 copies, then bilinear-gather 1024 BEV points from LDS,
// accumulating over the 6 cameras in registers. d_out written exactly once.
// ---------------------------------------------------------------------------
__global__ __launch_bounds__(THREADS)
void FeatureIPM_gather_kernel(const float* __restrict__ feats,
                              const float* __restrict__ proj,
                              const float* __restrict__ intr,
                              const float* __restrict__ c2c,
                              float* __restrict__ out,
                              int use_ws) {
  extern __shared__ float slab[];          // SLAB_FLOATS
  const int tid  = threadIdx.x;
  const int pblk = blockIdx.x;
  const int c0   = blockIdx.y * CH;
  const int b    = blockIdx.z;

  int   pt[PPT];
  float acc[PPT][CH];
#pragma unroll
  for (int q = 0; q < PPT; ++q) {
    pt[q] = pblk * PTS_PER_BLK + q * THREADS + tid;
#pragma unroll
    for (int cc = 0; cc < CH; ++cc) acc[q][cc] = 0.0f;
  }

  for (int n = 0; n < NCAM; ++n) {
    // ---- stage slab: channels [c0, c0+CH) are contiguous in feats ----
    const float4* src = (const float4*)(feats +
        ((size_t)(b * NCAM + n) * CHN + c0) * HFWF);
    float4* dst = (float4*)slab;
    const int CHUNKS = SLAB_FLOATS / 4;    // 9216 x 16B
    for (int i = tid; i < CHUNKS; i += THREADS) {
#if defined(USE_ASYNC_LDS)
      __builtin_amdgcn_global_load_async_to_lds_b128(
          ASYNC_GPTR(src + i), ASYNC_SPTR(dst + i), 0, 0);
#else
      dst[i] = src[i];
#endif
    }
    wait_async_zero();
    __syncthreads();

    const float* K = intr + (size_t)(b * NCAM + n) * 9;
    const float* M = c2c  + (size_t)(b * NCAM + n) * 16;

#pragma unroll
    for (int q = 0; q < PPT; ++q) {
      const int p = pt[q];
      if (p < NPTS) {
        float ix, iy;
        if (use_ws) {
          const float* pr = proj + ((size_t)(b * NCAM + n) * NPTS + p) * 2;
          ix = pr[0]; iy = pr[1];
        } else {
          project_point(K, M, p, ix, iy);
        }
        float x0f = floorf(ix), y0f = floorf(iy);
        float fx = ix - x0f,  fy = iy - y0f;
        int x0 = (int)x0f, y0 = (int)y0f;
        int x1 = x0 + 1,   y1 = y0 + 1;
        bool mx0 = ((unsigned)x0 <= (unsigned)(WF - 1));
        bool mx1 = ((unsigned)x1 <= (unsigned)(WF - 1));
        bool my0 = ((unsigned)y0 <= (unsigned)(HF - 1));
        bool my1 = ((unsigned)y1 <= (unsigned)(HF - 1));
        float w00 = (1.0f - fx) * (1.0f - fy) * ((mx0 && my0) ? 1.0f : 0.0f);
        float w01 = fx * (1.0f - fy)          * ((mx1 && my0) ? 1.0f : 0.0f);
        float w10 = (1.0f - fx) * fy          * ((mx0 && my1) ? 1.0f : 0.0f);
        float w11 = fx * fy                   * ((mx1 && my1) ? 1.0f : 0.0f);
        int xi0 = x0 < 0 ? 0 : (x0 > WF - 1 ? WF - 1 : x0);
        int xi1 = x1 < 0 ? 0 : (x1 > WF - 1 ? WF - 1 : x1);
        int yi0 = y0 < 0 ? 0 : (y0 > HF - 1 ? HF - 1 : y0);
        int yi1 = y1 < 0 ? 0 : (y1 > HF - 1 ? HF - 1 : y1);
        int l00 = yi0 * WF + xi0;
        int l01 = yi0 * WF + xi1;
        int l10 = yi1 * WF + xi0;
        int l11 = yi1 * WF + xi1;
#pragma unroll
        for (int cc = 0; cc < CH; ++cc) {
          const float* Lc = slab + cc * HFWF;
          float s = w00 * Lc[l00];
          s = fmaf(w01, Lc[l01], s);
          s = fmaf(w10, Lc[l10], s);
          s = fmaf(w11, Lc[l11], s);
          acc[q][cc] += s;
        }
      }
    }
    __syncthreads();   // slab reused next camera
  }

  const float inv_n = 1.0f / (float)NCAM;
#pragma unroll
  for (int q = 0; q < PPT; ++q) {
    const int p = pt[q];
    if (p < NPTS) {
#pragma unroll
      for (int cc = 0; cc < CH; ++cc) {
        out[((size_t)b * CHN + (c0 + cc)) * NPTS + p] = acc[q][cc] * inv_n;
      }
    }
  }
}

// ---------------------------------------------------------------------------
extern "C" void kernel_launch(void* const* d_in, const int* in_sizes, int n_in,
                              void* d_out, int out_size, void* d_ws, size_t ws_size,
                              hipStream_t stream) {
  const float* feats = (const float*)d_in[0];   // [2,6,256,48,96] f32
  const float* intr  = (const float*)d_in[1];   // [2,6,3,3] f32
  const float* c2c   = (const float*)d_in[2];   // [2,6,4,4] f32
  float* out = (float*)d_out;                   // [2,256,188,126] f32
  float* proj = (float*)d_ws;                   // [2,6,23688,2] f32 coords

  const size_t ws_needed = (size_t)BATCH * NCAM * NPTS * 2 * sizeof(float);
  const int use_ws = (ws_size >= ws_needed) ? 1 : 0;

  if (use_ws) {
    dim3 grid((NPTS + 127) / 128, NCAM, BATCH);   // 186 x 6 x 2
    FeatureIPM_proj_kernel<<<grid, THREADS, 0, stream>>>(intr, c2c, proj);
  }

  dim3 ggrid((NPTS + PTS_PER_BLK - 1) / PTS_PER_BLK, CHN / CH, BATCH); // 24 x 32 x 2
  FeatureIPM_gather_kernel<<<ggrid, THREADS, SLAB_BYTES, stream>>>(
      feats, proj, intr, c2c, out, use_ws);
}